// HTNE_1176821039722
// MI455X (gfx1250) — compile-verified
//
#include <hip/hip_runtime.h>
#include <hip/hip_bf16.h>
#include <math.h>

// ---------------- configuration ----------------
#define WPB   8              // waves per block (wave32)
#define TPB   (WPB * 32)     // 256 threads
#define ITERS 8              // edges per wave
#define NH    5              // history length H
#define DD    128            // embedding dim D
#define NROWS 7              // s, t, h0..h4

// Detect gfx1250 async global->LDS builtins (guarded so the file always builds).
#if defined(__HIP_DEVICE_COMPILE__) && defined(__has_builtin)
#  if __has_builtin(__builtin_amdgcn_global_load_async_to_lds_b128) && \
      __has_builtin(__builtin_amdgcn_s_wait_asynccnt)
#    define HAVE_ASYNC 1
#  endif
#endif
#ifndef HAVE_ASYNC
#  define HAVE_ASYNC 0
#endif

#if HAVE_ASYNC
// Builtin prototype (from hipcc diagnostic): (v4i as1*, v4i as3*, Ii, Ii)
typedef int v4i __attribute__((vector_size(4 * sizeof(int))));
typedef __attribute__((address_space(1))) v4i gv4i;   // global
typedef __attribute__((address_space(3))) v4i lv4i;   // LDS
#endif

__global__ __launch_bounds__(TPB)
void htne_loss_kernel(const int*   __restrict__ sArr,
                      const int*   __restrict__ tArr,
                      const float* __restrict__ edgeT,
                      const int*   __restrict__ hIdx,
                      const float* __restrict__ hT,
                      const float* __restrict__ hM,
                      const float* __restrict__ signp,
                      const float* __restrict__ emb,
                      const float* __restrict__ deltaTab,
                      float*       __restrict__ out,
                      int n)
{
  // Group partial sums: [wave][edge][reduction r][lane-group g], 16B rows.
  __shared__ float redpart[WPB][ITERS][6][4];                 // 6 KB
#if HAVE_ASYNC
  __shared__ float smem[WPB][2][NROWS][DD];                   // 56 KB, 2-stage gather
#endif
  const int lane = threadIdx.x & 31;
  const int wv   = threadIdx.x >> 5;
  const long long base = ((long long)blockIdx.x * WPB + wv) * (long long)ITERS;
  if (base >= n) return;

#if HAVE_ASYNC
  // Issue 7 async B128 gathers (one full 512B row per instruction, 16B/lane,
  // perfectly coalesced) into LDS stage (i&1). Tracked by ASYNCcnt.
  auto issue = [&](int i) {
    const long long e  = base + i;
    const int       st = i & 1;
    long long ridx[NROWS];
    ridx[0] = sArr[e];
    ridx[1] = tArr[e];
#pragma unroll
    for (int h = 0; h < NH; ++h) ridx[2 + h] = hIdx[e * NH + h];
#pragma unroll
    for (int r = 0; r < NROWS; ++r) {
      const float* g = emb + ridx[r] * (long long)DD + lane * 4;
      float*       l = &smem[wv][st][r][lane * 4];
      __builtin_amdgcn_global_load_async_to_lds_b128((gv4i*)g, (lv4i*)l, 0, 0);
    }
  };
#endif

  // Distances + 3-stage butterfly (groups of 8 lanes), park 6x4 partials in LDS.
  auto gather_reduce = [&](int i) {
    const long long e  = base + i;
    const int       st = i & 1;
    (void)st; (void)e;
#if HAVE_ASYNC
    const float4 sv = *(const float4*)&smem[wv][st][0][lane * 4];
    const float4 tv = *(const float4*)&smem[wv][st][1][lane * 4];
#else
    const float4 sv = *(const float4*)(emb + (long long)sArr[e] * DD + lane * 4);
    const float4 tv = *(const float4*)(emb + (long long)tArr[e] * DD + lane * 4);
#endif
    float red[6];
    {
      const float dx = sv.x - tv.x, dy = sv.y - tv.y,
                  dz = sv.z - tv.z, dw = sv.w - tv.w;
      red[0] = dx * dx + dy * dy + dz * dz + dw * dw;
    }
#pragma unroll
    for (int h = 0; h < NH; ++h) {
#if HAVE_ASYNC
      const float4 hv = *(const float4*)&smem[wv][st][2 + h][lane * 4];
#else
      const float4 hv =
          *(const float4*)(emb + (long long)hIdx[e * NH + h] * DD + lane * 4);
#endif
      const float dx = sv.x - hv.x, dy = sv.y - hv.y,
                  dz = sv.z - hv.z, dw = sv.w - hv.w;
      red[1 + h] = dx * dx + dy * dy + dz * dz + dw * dw;
    }

    // 3 butterfly stages: lane l ends with the sum over its aligned 8-lane group
    // (i.e. elements [32g .. 32g+31] of D for g = l>>3).
#pragma unroll
    for (int m = 1; m <= 4; m <<= 1) {
#pragma unroll
      for (int r = 0; r < 6; ++r) red[r] += __shfl_xor(red[r], m, 32);
    }

    // Lane l = 8g + j (j<6) stores reduction j's group-g partial: one ds_store.
    const int j = lane & 7;
    const int g = lane >> 3;
    float v = red[0];
    v = (j == 1) ? red[1] : v;
    v = (j == 2) ? red[2] : v;
    v = (j == 3) ? red[3] : v;
    v = (j == 4) ? red[4] : v;
    v = (j == 5) ? red[5] : v;
    if (j < 6) redpart[wv][i][j][g] = v;
  };

  // ---------------- pipelined gather loop ----------------
#if HAVE_ASYNC
  issue(0);
#pragma unroll
  for (int i = 0; i < ITERS; ++i) {
    if (base + i >= n) break;
    const bool more = (i + 1 < ITERS) && (base + i + 1 < n);
    if (more) {
      issue(i + 1);
      __builtin_amdgcn_s_wait_asynccnt(NROWS);  // prev stage's 7 loads done (in-order)
    } else {
      __builtin_amdgcn_s_wait_asynccnt(0);
    }
    asm volatile("" ::: "memory");  // keep LDS reads inside this iteration
    gather_reduce(i);
    asm volatile("" ::: "memory");
  }
#else
#pragma unroll
  for (int i = 0; i < ITERS; ++i) {
    if (base + i >= n) break;
    gather_reduce(i);
  }
#endif

  // ---------------- lane-parallel epilogue: lane l owns edge base+l ----------
  if (lane < ITERS && base + lane < n) {
    const long long e = base + lane;
    float red[6];
#pragma unroll
    for (int r = 0; r < 6; ++r) {
      const float4 p = *(const float4*)&redpart[wv][lane][r][0];
      red[r] = (p.x + p.y) + (p.z + p.w);
    }
    const float p_mu = -red[0];
    float alpha[NH];
    float amax = -red[1];
#pragma unroll
    for (int h = 0; h < NH; ++h) {
      alpha[h] = -red[1 + h];
      amax = fmaxf(amax, alpha[h]);
    }
    float w[NH], den = 0.f;
#pragma unroll
    for (int h = 0; h < NH; ++h) {
      w[h] = __expf(alpha[h] - amax);
      den += w[h];
    }
    const float inv_den = 1.f / den;
    const float delta   = deltaTab[sArr[e]];
    const float et      = edgeT[e];
    float pl = p_mu;
#pragma unroll
    for (int h = 0; h < NH; ++h) {
      const float dt = et - hT[e * NH + h];
      pl += (w[h] * inv_den) * alpha[h] * __expf(-delta * dt) * hM[e * NH + h];
    }
    const float x    = signp[0] * pl;
    const float loss = fmaxf(-x, 0.f) + log1pf(__expf(-fabsf(x)));  // -log_sigmoid
    out[e] = loss;
  }
}

extern "C" void kernel_launch(void* const* d_in, const int* in_sizes, int n_in,
                              void* d_out, int out_size, void* d_ws, size_t ws_size,
                              hipStream_t stream) {
  const int*   s    = (const int*)  d_in[0];
  const int*   t    = (const int*)  d_in[1];
  const float* et   = (const float*)d_in[2];
  const int*   hs   = (const int*)  d_in[3];
  const float* hst  = (const float*)d_in[4];
  const float* hsm  = (const float*)d_in[5];
  const float* sg   = (const float*)d_in[6];
  const float* emb  = (const float*)d_in[7];
  const float* dtab = (const float*)d_in[8];
  float* out = (float*)d_out;

  const int n      = in_sizes[0];           // B edges
  const int epb    = WPB * ITERS;           // edges per block
  const int blocks = (n + epb - 1) / epb;

  hipLaunchKernelGGL(htne_loss_kernel, dim3(blocks), dim3(TPB), 0, stream,
                     s, t, et, hs, hst, hsm, sg, emb, dtab, out, n);
}